// FPFHLAttn_67147518705772
// MI455X (gfx1250) — compile-verified
//
#include <hip/hip_runtime.h>
#include <hip/hip_bf16.h>

// ---------------- problem constants ----------------
#define BATCH 2
#define NPTS  8192
#define SNODE 256
#define NSAMP 32
#define DFEAT 384
#define NFPFH 33
#define FPAD  36          // 33 padded to multiple of 4 with zeros (distance-invariant)
#define EPS_SK 0.1f
#define INV_EPS 10.0f
#define SK_ITERS 100
#define INV_SQRT_D 0.05103103630798287f   // 1/sqrt(384)

typedef __attribute__((ext_vector_type(2))) float v2f;
typedef __attribute__((ext_vector_type(8))) float v8f;
typedef __attribute__((ext_vector_type(4))) int   v4i;

#define AS1 __attribute__((address_space(1)))
#define AS3 __attribute__((address_space(3)))

// =====================================================================
// Kernel 1: 32-NN of each downsampled node among 8192 points.
// One wave per node; squared distances staged in LDS; 32 argmin passes
// with lowest-index tie-break (matches jax.lax.top_k ordering).
// =====================================================================
__global__ __launch_bounds__(32) void knn_kernel(
    const float* __restrict__ down_xyz, const float* __restrict__ xyz,
    int* __restrict__ idx_out) {
  __shared__ float dist[NPTS];
  const int node = blockIdx.x;            // 0..511  (= bb*256 + sp)
  const int bb   = node >> 8;
  const int lane = threadIdx.x;

  const float cx = down_xyz[node * 3 + 0];
  const float cy = down_xyz[node * 3 + 1];
  const float cz = down_xyz[node * 3 + 2];
  const float* X = xyz + (size_t)bb * NPTS * 3;

  for (int i = lane; i < NPTS; i += 32) {
    float dx = X[i * 3 + 0] - cx;
    float dy = X[i * 3 + 1] - cy;
    float dz = X[i * 3 + 2] - cz;
    dist[i] = dx * dx + dy * dy + dz * dz;   // sqrt is monotone: same order/ties
  }
  __syncthreads();

  for (int kk = 0; kk < NSAMP; ++kk) {
    float bv = 3.4e38f; int bi = 0x7fffffff;
    for (int i = lane; i < NPTS; i += 32) {
      float v = dist[i];
      if (v < bv || (v == bv && i < bi)) { bv = v; bi = i; }
    }
    #pragma unroll
    for (int off = 16; off > 0; off >>= 1) {
      float ov = __shfl_xor(bv, off, 32);
      int   oi = __shfl_xor(bi, off, 32);
      if (ov < bv || (ov == bv && oi < bi)) { bv = ov; bi = oi; }
    }
    if (lane == 0) { idx_out[node * NSAMP + kk] = bi; dist[bi] = 3.4e38f; }
    __syncthreads();
  }
}

// --------- wave32 log-domain Sinkhorn on a symmetric 32x32 cost ---------
// Lane i owns row i in registers; g_j / f_i broadcast via shuffles.
// Overwrites the LDS cost with the transport plan gamma.
__device__ __forceinline__ void sinkhorn32(float* Cp, int lane) {
  float Crow[32];
  #pragma unroll
  for (int j = 0; j < 32; ++j) Crow[j] = Cp[lane * 32 + j];
  const float logp = -3.4657359027997265f;   // log(1/32)
  float f = 0.f, g = 0.f;
  for (int it = 0; it < SK_ITERS; ++it) {
    // f-update: lane i does LSE over j of (-C[i][j] + f_i + g_j)/eps
    float t[32];
    float m = -3.4e38f;
    #pragma unroll
    for (int j = 0; j < 32; ++j) {
      float gj = __shfl(g, j, 32);
      t[j] = (-Crow[j] + f + gj) * INV_EPS;
      m = fmaxf(m, t[j]);
    }
    float s = 0.f;
    #pragma unroll
    for (int j = 0; j < 32; ++j) s += __expf(t[j] - m);
    f = f + EPS_SK * (logp - (m + __logf(s)));
    // g-update: C symmetric, so lane j's own row supplies C[i][j]
    float m2 = -3.4e38f;
    #pragma unroll
    for (int i = 0; i < 32; ++i) {
      float fi = __shfl(f, i, 32);
      t[i] = (-Crow[i] + fi + g) * INV_EPS;
      m2 = fmaxf(m2, t[i]);
    }
    float s2 = 0.f;
    #pragma unroll
    for (int i = 0; i < 32; ++i) s2 += __expf(t[i] - m2);
    g = g + EPS_SK * (logp - (m2 + __logf(s2)));
  }
  #pragma unroll
  for (int j = 0; j < 32; ++j) {
    float gj = __shfl(g, j, 32);
    Cp[lane * 32 + j] = __expf((-Crow[j] + f + gj) * INV_EPS);
  }
}

// =====================================================================
// Kernel 2: one workgroup (8 waves) per patch. Async-gathers features
// into LDS, does both Gram matrices with V_WMMA_F32_16X16X4_F32 (fully
// specialized, uniform loops), dual register-resident Sinkhorn, attn
// product/normalize, then attn @ feats with WMMA.
// =====================================================================
__global__ __launch_bounds__(256) void patch_attn_kernel(
    const float* __restrict__ xyz, const float* __restrict__ xyz_feats,
    const float* __restrict__ xyz_fpfhs, const int* __restrict__ knn_idx,
    float* __restrict__ flat_xyz, float* __restrict__ flat_feats) {
  __shared__ __align__(16) float feats[NSAMP * DFEAT];   // 48 KB
  __shared__ __align__(16) float fpfh[NSAMP * FPAD];     // 4.5 KB (zero-pad 33..35)
  __shared__ float vc[NSAMP * NSAMP];      // feat cost -> gamma_v
  __shared__ float gc[NSAMP * NSAMP];      // fpfh cost -> gamma_g
  __shared__ float attnS[NSAMP * NSAMP];   // normalized attention
  __shared__ float nrmF[NSAMP], nrmG[NSAMP];
  __shared__ int   pidx[NSAMP];

  const int pid  = blockIdx.x;             // 0..511
  const int bb   = pid >> 8;
  const int sp   = pid & 255;
  const int tid  = threadIdx.x;
  const int wave = tid >> 5;
  const int lane = tid & 31;
  const int half = lane >> 4;
  const int r    = lane & 15;

  if (tid < NSAMP) pidx[tid] = knn_idx[pid * NSAMP + tid];
  __syncthreads();

  // ---- gather patch features: async Global->LDS (ASYNCcnt path) ----
#if __has_builtin(__builtin_amdgcn_global_load_async_to_lds_b128)
  for (int ch = tid; ch < NSAMP * (DFEAT / 4); ch += 256) {   // 16B chunks
    const int row = ch / (DFEAT / 4);
    const int cf  = (ch - row * (DFEAT / 4)) * 4;             // float offset in row
    AS1 v4i* g = (AS1 v4i*)(xyz_feats +
        ((size_t)bb * NPTS + pidx[row]) * DFEAT + cf);
    AS3 v4i* l = (AS3 v4i*)(&feats[row * DFEAT + cf]);
    __builtin_amdgcn_global_load_async_to_lds_b128(g, l, 0, 0);
  }
#if __has_builtin(__builtin_amdgcn_s_wait_asynccnt)
  __builtin_amdgcn_s_wait_asynccnt(0);
#else
  asm volatile("s_wait_asynccnt 0x0" ::: "memory");
#endif
#else
  for (int i = tid; i < NSAMP * DFEAT; i += 256) {
    int row = i / DFEAT, col = i - row * DFEAT;
    feats[i] = xyz_feats[((size_t)bb * NPTS + pidx[row]) * DFEAT + col];
  }
#endif
  for (int i = tid; i < NSAMP * FPAD; i += 256) {
    int row = i / FPAD, col = i - row * FPAD;
    fpfh[i] = (col < NFPFH)
                  ? xyz_fpfhs[((size_t)bb * NPTS + pidx[row]) * NFPFH + col]
                  : 0.f;
  }
  for (int i = tid; i < NSAMP * 3; i += 256) {
    int row = i / 3, c = i - row * 3;
    flat_xyz[((size_t)(bb * SNODE + sp) * NSAMP + row) * 3 + c] =
        xyz[((size_t)bb * NPTS + pidx[row]) * 3 + c];
  }
  __syncthreads();

  // ---- squared norms (waves 0/1) ----
  if (wave == 0) {
    float s = 0.f;
    for (int c = 0; c < DFEAT; ++c) { float v = feats[lane * DFEAT + c]; s += v * v; }
    nrmF[lane] = s;
  } else if (wave == 1) {
    float s = 0.f;
    for (int c = 0; c < FPAD; ++c) { float v = fpfh[lane * FPAD + c]; s += v * v; }
    nrmG[lane] = s;
  }
  __syncthreads();

  // ---- Gram matrices via f32 WMMA; compile-time specialized per path ----
  if (wave < 4) {                    // feature Gram, K = 384
    const int mo = (wave & 1) << 4;
    const int no = ((wave >> 1) & 1) << 4;
    v8f acc = {};
    #pragma unroll 8
    for (int k = 0; k < DFEAT; k += 4) {
      const int kb = k + (half << 1);
      v2f a = *reinterpret_cast<const v2f*>(&feats[(mo + r) * DFEAT + kb]);
      v2f b = *reinterpret_cast<const v2f*>(&feats[(no + r) * DFEAT + kb]);
      acc = __builtin_amdgcn_wmma_f32_16x16x4_f32(false, a, false, b,
                                                  (short)0, acc, false, false);
    }
    #pragma unroll
    for (int v = 0; v < 8; ++v) {
      int m = mo + v + (half << 3);
      int n = no + r;
      float sq = nrmF[m] + nrmF[n] - 2.0f * acc[v];
      vc[m * NSAMP + n] = (sq > 0.f) ? sqrtf(sq) * INV_SQRT_D : 0.f;
    }
  } else {                           // FPFH Gram, K = 36 (fully unrolled)
    const int w  = wave - 4;
    const int mo = (w & 1) << 4;
    const int no = ((w >> 1) & 1) << 4;
    v8f acc = {};
    #pragma unroll
    for (int k = 0; k < FPAD; k += 4) {
      const int kb = k + (half << 1);
      v2f a = *reinterpret_cast<const v2f*>(&fpfh[(mo + r) * FPAD + kb]);
      v2f b = *reinterpret_cast<const v2f*>(&fpfh[(no + r) * FPAD + kb]);
      acc = __builtin_amdgcn_wmma_f32_16x16x4_f32(false, a, false, b,
                                                  (short)0, acc, false, false);
    }
    #pragma unroll
    for (int v = 0; v < 8; ++v) {
      int m = mo + v + (half << 3);
      int n = no + r;
      float sq = nrmG[m] + nrmG[n] - 2.0f * acc[v];
      gc[m * NSAMP + n] = (sq > 0.f) ? sqrtf(sq) * INV_SQRT_D : 0.f;
    }
  }
  __syncthreads();

  // ---- two independent Sinkhorns (uniform marginals 1/32 for both) ----
  if (wave == 0) sinkhorn32(vc, lane);
  else if (wave == 1) sinkhorn32(gc, lane);
  __syncthreads();

  // ---- attn = rownorm(gamma_v * gamma_g) ----
  if (wave == 0) {
    float row[32]; float s = 0.f;
    #pragma unroll
    for (int j = 0; j < 32; ++j) {
      float a = vc[lane * 32 + j] * gc[lane * 32 + j];
      row[j] = a; s += a;
    }
    float inv = 1.0f / s;
    #pragma unroll
    for (int j = 0; j < 32; ++j) attnS[lane * 32 + j] = row[j] * inv;
  }
  __syncthreads();

  // ---- attn_feats = attn(32x32) @ feats(32x384) via f32 WMMA ----
  // 48 tiles (2 Mtiles x 24 Ntiles), 6 per wave; uniform trip count.
  float* outF = flat_feats + (size_t)(bb * SNODE + sp) * NSAMP * DFEAT;
  for (int t2 = 0; t2 < 6; ++t2) {
    const int tile = wave + (t2 << 3);
    const int mo = (tile & 1) << 4;
    const int no = (tile >> 1) << 4;
    v8f acc = {};
    #pragma unroll
    for (int k = 0; k < NSAMP; k += 4) {
      const int kb = k + (half << 1);
      v2f a = *reinterpret_cast<const v2f*>(&attnS[(mo + r) * NSAMP + kb]);
      v2f b;
      b.x = feats[kb * DFEAT + no + r];
      b.y = feats[(kb + 1) * DFEAT + no + r];
      acc = __builtin_amdgcn_wmma_f32_16x16x4_f32(false, a, false, b,
                                                  (short)0, acc, false, false);
    }
    #pragma unroll
    for (int v = 0; v < 8; ++v) {
      int m = mo + v + (half << 3);
      int n = no + r;
      outF[(size_t)m * DFEAT + n] = acc[v];
    }
  }
}

// =====================================================================
// Kernel 3: nearest patch-point for every original point + fused output.
// 64 points per block, candidates tiled through LDS, 4 threads/point.
// =====================================================================
__global__ __launch_bounds__(256) void nn_out_kernel(
    const float* __restrict__ xyz, const float* __restrict__ xyz_feats,
    const float* __restrict__ flat_xyz, const float* __restrict__ flat_feats,
    float* __restrict__ out) {
  const int PTS = 64, TILE = 1024;
  __shared__ float cand[TILE * 3];
  __shared__ float rv[256];
  __shared__ int   ri[256];
  __shared__ int   nn[PTS];

  const int blk   = blockIdx.x;           // 0..255
  const int bb    = blk >> 7;
  const int pbase = (blk & 127) * PTS;
  const int tid   = threadIdx.x;
  const int pt    = tid >> 2, sub = tid & 3;
  const int p     = pbase + pt;

  const float px = xyz[((size_t)bb * NPTS + p) * 3 + 0];
  const float py = xyz[((size_t)bb * NPTS + p) * 3 + 1];
  const float pz = xyz[((size_t)bb * NPTS + p) * 3 + 2];

  float bv = 3.4e38f; int bi = 0x7fffffff;
  const float* FX = flat_xyz + (size_t)bb * NPTS * 3;

  for (int t = 0; t < NPTS; t += TILE) {
    __syncthreads();
    for (int i = tid; i < TILE * 3; i += 256) cand[i] = FX[t * 3 + i];
    if (t + TILE < NPTS)
      __builtin_prefetch(&FX[(t + TILE) * 3 + (tid * 12)], 0, 0);  // global_prefetch_b8
    __syncthreads();
    for (int c = sub; c < TILE; c += 4) {
      float dx = cand[c * 3 + 0] - px;
      float dy = cand[c * 3 + 1] - py;
      float dz = cand[c * 3 + 2] - pz;
      float d2 = dx * dx + dy * dy + dz * dz;
      int gi = t + c;
      if (d2 < bv || (d2 == bv && gi < bi)) { bv = d2; bi = gi; }
    }
  }
  rv[tid] = bv; ri[tid] = bi;
  __syncthreads();
  if (sub == 0) {
    #pragma unroll
    for (int s2 = 1; s2 < 4; ++s2) {
      float ov = rv[tid + s2]; int oi = ri[tid + s2];
      if (ov < bv || (ov == bv && oi < bi)) { bv = ov; bi = oi; }
    }
    nn[pt] = bi;
  }
  __syncthreads();

  const float* XF = xyz_feats  + ((size_t)bb * NPTS + pbase) * DFEAT;
  const float* FF = flat_feats + (size_t)bb * NPTS * DFEAT;
  float* O        = out        + ((size_t)bb * NPTS + pbase) * DFEAT;
  for (int e = tid; e < PTS * DFEAT; e += 256) {
    int pp = e / DFEAT, c = e - pp * DFEAT;
    O[e] = 0.5f * (XF[e] + FF[(size_t)nn[pp] * DFEAT + c]);
  }
}

// Kernel 4: tuple's second output = xyz_fpfhs passthrough
__global__ void copy_fpfh_kernel(const float* __restrict__ src,
                                 float* __restrict__ dst, int n) {
  int i = blockIdx.x * blockDim.x + threadIdx.x;
  if (i < n) dst[i] = src[i];
}

extern "C" void kernel_launch(void* const* d_in, const int* in_sizes, int n_in,
                              void* d_out, int out_size, void* d_ws, size_t ws_size,
                              hipStream_t stream) {
  (void)in_sizes; (void)n_in; (void)out_size; (void)ws_size;
  const float* down_xyz   = (const float*)d_in[0];   // [2,256,3]
  const float* xyz        = (const float*)d_in[1];   // [2,8192,3]
  // d_in[2] = down_fpfhs: unused by the reference
  const float* xyz_fpfhs  = (const float*)d_in[3];   // [2,8192,33]
  const float* xyz_feats  = (const float*)d_in[4];   // [2,8192,384]

  float* out      = (float*)d_out;                              // [2,8192,384]
  float* out_fpfh = out + (size_t)BATCH * NPTS * DFEAT;         // [2,8192,33]

  // workspace layout
  char* ws = (char*)d_ws;
  int*   knn_idx    = (int*)ws;                                        // 16384 ints
  float* flat_xyz   = (float*)(ws + (size_t)BATCH * SNODE * NSAMP * sizeof(int));
  float* flat_feats = flat_xyz + (size_t)BATCH * SNODE * NSAMP * 3;    // [2,8192,384]

  knn_kernel<<<BATCH * SNODE, 32, 0, stream>>>(down_xyz, xyz, knn_idx);
  patch_attn_kernel<<<BATCH * SNODE, 256, 0, stream>>>(
      xyz, xyz_feats, xyz_fpfhs, knn_idx, flat_xyz, flat_feats);
  nn_out_kernel<<<BATCH * (NPTS / 64), 256, 0, stream>>>(
      xyz, xyz_feats, flat_xyz, flat_feats, out);
  const int nf = BATCH * NPTS * NFPFH;
  copy_fpfh_kernel<<<(nf + 255) / 256, 256, 0, stream>>>(xyz_fpfhs, out_fpfh, nf);
}